// GraphConvModule_86260123174004
// MI455X (gfx1250) — compile-verified
//
#include <hip/hip_runtime.h>

// Edge-conditioned graph conv, fused as one WMMA GEMM:
//   out[e,o] = sum_k z[e,k] * Wbig[k,o],  K = 112
//   z[e, f*16+i] = edgefeats[e,f]*g[e,i]   (f<6)     [outer product]
//   z[e, 96+i]   = g[e,i]                            [bias row]
//   Wbig[f*16+i, o] = fnet_W[f, i*32+o];  Wbig[96+i, o] = fnet_b[i*32+o]
// then segment-mean over sorted idxd.

#define CIN   16
#define COUT  32
#define EFDIM 6
#define KDIM  112            // 6*16 + 16 bias rows
#define KCH   (KDIM / 4)     // 28 chunks of K=4 for V_WMMA_F32_16X16X4_F32

typedef float v2f __attribute__((ext_vector_type(2)));
typedef float v8f __attribute__((ext_vector_type(8)));

// k is a compile-time constant at every use site (fully unrolled loop),
// so array indices stay constant -> registers, no scratch.
#define ZVAL(k) (((k) < 96) ? (efr[(k) >> 4] * g[(k) & 15]) : (g[(k) - 96]))

__global__ __launch_bounds__(256) void ecc_zero_kernel(float* __restrict__ out,
                                                       int* __restrict__ counts,
                                                       int out_n, int node_n) {
    int i = blockIdx.x * blockDim.x + threadIdx.x;
    if (i < out_n) out[i] = 0.0f;
    if (i < node_n) counts[i] = 0;
}

__global__ __launch_bounds__(256) void ecc_count_kernel(const int* __restrict__ idxd,
                                                        int* __restrict__ counts, int n) {
    int i = blockIdx.x * blockDim.x + threadIdx.x;
    if (i < n) atomicAdd(&counts[idxd[i]], 1);
}

__global__ __launch_bounds__(256) void ecc_main_kernel(
    const float* __restrict__ input,    // [N, 16]
    const float* __restrict__ edgef,    // [E, 6]
    const float* __restrict__ fW,       // [6, 512]
    const float* __restrict__ fb,       // [512]
    const int*   __restrict__ idxn,     // [E]
    const int*   __restrict__ idxd,     // [E] sorted
    float*       __restrict__ out,      // [N, 32] accumulated via atomics
    int n_tiles) {
    // B matrix (constant across all edges) staged in LDS: 112x32 f32 = 14336 B.
    __shared__ float Wb[KDIM * COUT];
    for (int t = threadIdx.x; t < KDIM * COUT; t += blockDim.x) {
        int k = t >> 5, o = t & 31;
        float v;
        if (k < 96) v = fW[(k >> 4) * (CIN * COUT) + (k & 15) * COUT + o];
        else        v = fb[(k - 96) * COUT + o];
        Wb[t] = v;
    }
    __syncthreads();

    const int lane = threadIdx.x & 31;
    const int wave = threadIdx.x >> 5;
    const int half = lane >> 4;      // lane group: 0 -> K pair {0,1}, 1 -> {2,3}
    const int nlo  = lane & 15;      // A row index / B,N column index
    const int wavesPerBlock = blockDim.x >> 5;
    const int gwave   = blockIdx.x * wavesPerBlock + wave;
    const int gstride = gridDim.x * wavesPerBlock;
    const int koff = half * (2 * COUT); // LDS offset for this lane group's K rows

    for (int tile = gwave; tile < n_tiles; tile += gstride) {
        const int ebase = tile * 16;
        const int erow  = ebase + nlo;     // lanes L and L+16 share the same edge row

        // Gather source-node features (input table is L2-resident: 3.2 MB).
        const int src = idxn[erow];
        float g[16];
        {
            const float4* gp = reinterpret_cast<const float4*>(input + (size_t)src * CIN);
            float4 t0 = gp[0], t1 = gp[1], t2 = gp[2], t3 = gp[3];
            g[0]=t0.x; g[1]=t0.y; g[2]=t0.z; g[3]=t0.w;
            g[4]=t1.x; g[5]=t1.y; g[6]=t1.z; g[7]=t1.w;
            g[8]=t2.x; g[9]=t2.y; g[10]=t2.z; g[11]=t2.w;
            g[12]=t3.x; g[13]=t3.y; g[14]=t3.z; g[15]=t3.w;
        }
        float efr[EFDIM];
        {
            const float2* ep = reinterpret_cast<const float2*>(edgef + (size_t)erow * EFDIM);
            float2 e01 = ep[0], e23 = ep[1], e45 = ep[2];
            efr[0]=e01.x; efr[1]=e01.y; efr[2]=e23.x;
            efr[3]=e23.y; efr[4]=e45.x; efr[5]=e45.y;
        }

        v8f acc_lo = {};  // output channels 0..15
        v8f acc_hi = {};  // output channels 16..31

        // 28 chunks of K=4: A built on the fly (cndmask on lane group), B from LDS.
        #pragma unroll
        for (int kc = 0; kc < KCH; ++kc) {
            const int kbase = 4 * kc;
            // A fragment: lanes 0-15 hold K = kbase+0/+1, lanes 16-31 hold K = kbase+2/+3.
            float a0 = half ? ZVAL(kbase + 2) : ZVAL(kbase + 0);
            float a1 = half ? ZVAL(kbase + 3) : ZVAL(kbase + 1);
            v2f A = {a0, a1};
            // B fragment mirrors A: vgpr0 = row K=kbase(+2*half), vgpr1 = next row.
            const int bb = kbase * COUT + koff + nlo;
            v2f Bl = { Wb[bb],            Wb[bb + COUT] };
            v2f Bh = { Wb[bb + 16],       Wb[bb + COUT + 16] };
            acc_lo = __builtin_amdgcn_wmma_f32_16x16x4_f32(
                false, A, false, Bl, (short)0, acc_lo, false, false);
            acc_hi = __builtin_amdgcn_wmma_f32_16x16x4_f32(
                false, A, false, Bh, (short)0, acc_hi, false, false);
        }

        // D layout: vgpr r, lanes 0-15 -> edge row r, lanes 16-31 -> edge row r+8.
        // idxd is sorted -> merge equal-destination runs before atomics (~8x fewer).
        const int* dptr = idxd + ebase + 8 * half;
        int4 dA = *reinterpret_cast<const int4*>(dptr);
        int4 dB = *reinterpret_cast<const int4*>(dptr + 4);
        int   dd[8] = {dA.x, dA.y, dA.z, dA.w, dB.x, dB.y, dB.z, dB.w};
        float vlo[8], vhi[8];
        #pragma unroll
        for (int r = 0; r < 8; ++r) { vlo[r] = acc_lo[r]; vhi[r] = acc_hi[r]; }

        int cur = dd[0];
        float slo = vlo[0], shi = vhi[0];
        #pragma unroll
        for (int r = 1; r < 8; ++r) {
            if (dd[r] == cur) {
                slo += vlo[r]; shi += vhi[r];
            } else {
                atomicAdd(out + (size_t)cur * COUT + nlo,      slo);
                atomicAdd(out + (size_t)cur * COUT + nlo + 16, shi);
                cur = dd[r]; slo = vlo[r]; shi = vhi[r];
            }
        }
        atomicAdd(out + (size_t)cur * COUT + nlo,      slo);
        atomicAdd(out + (size_t)cur * COUT + nlo + 16, shi);
    }
}

// Scalar fallback for a (<16)-edge tail; not used for E = 640000 but kept for safety.
__global__ __launch_bounds__(256) void ecc_tail_kernel(
    const float* __restrict__ input, const float* __restrict__ edgef,
    const float* __restrict__ fW, const float* __restrict__ fb,
    const int* __restrict__ idxn, const int* __restrict__ idxd,
    float* __restrict__ out, int estart, int n_edges) {
    int t = blockIdx.x * blockDim.x + threadIdx.x;
    int e = estart + (t >> 5);
    int o = t & 31;
    if (e >= n_edges) return;
    const float* gp = input + (size_t)idxn[e] * CIN;
    const float* ep = edgef + (size_t)e * EFDIM;
    float s = 0.0f;
    for (int i = 0; i < CIN; ++i) {
        float w = fb[i * COUT + o];
        for (int f = 0; f < EFDIM; ++f)
            w += ep[f] * fW[f * (CIN * COUT) + i * COUT + o];
        s += gp[i] * w;
    }
    atomicAdd(out + (size_t)idxd[e] * COUT + o, s);
}

__global__ __launch_bounds__(256) void ecc_fin_kernel(float* __restrict__ out,
                                                      const int* __restrict__ counts,
                                                      int out_n) {
    int i = blockIdx.x * blockDim.x + threadIdx.x;
    if (i >= out_n) return;
    int c = counts[i >> 5];
    out[i] = (c > 0) ? (out[i] / (float)c) : 0.0f;
}

extern "C" void kernel_launch(void* const* d_in, const int* in_sizes, int n_in,
                              void* d_out, int out_size, void* d_ws, size_t ws_size,
                              hipStream_t stream) {
    const float* input = (const float*)d_in[0];
    const float* edgef = (const float*)d_in[1];
    const float* fW    = (const float*)d_in[2];
    const float* fb    = (const float*)d_in[3];
    const int*   idxn  = (const int*)d_in[4];
    const int*   idxd  = (const int*)d_in[5];
    float* out = (float*)d_out;

    const int n_edges = in_sizes[4];
    const int n_nodes = out_size / COUT;
    int* counts = (int*)d_ws;   // n_nodes ints of scratch

    // 1) zero accumulators (harness poisons d_out/d_ws).
    int zn = (out_size > n_nodes) ? out_size : n_nodes;
    ecc_zero_kernel<<<(zn + 255) / 256, 256, 0, stream>>>(out, counts, out_size, n_nodes);

    // 2) per-node degree.
    ecc_count_kernel<<<(n_edges + 255) / 256, 256, 0, stream>>>(idxd, counts, n_edges);

    // 3) fused filter-net + per-edge matvec as [E,112]x[112,32] WMMA GEMM + scatter-add.
    const int n_tiles = n_edges / 16;
    if (n_tiles > 0) {
        int blocks = (n_tiles + 7) / 8;       // 8 waves per 256-thread block
        if (blocks > 2500) blocks = 2500;     // grid-stride beyond this
        ecc_main_kernel<<<blocks, 256, 0, stream>>>(input, edgef, fW, fb, idxn, idxd,
                                                    out, n_tiles);
    }
    const int tail = n_edges - n_tiles * 16;
    if (tail > 0) {
        ecc_tail_kernel<<<(tail * COUT + 255) / 256, 256, 0, stream>>>(
            input, edgef, fW, fb, idxn, idxd, out, n_tiles * 16, n_edges);
    }

    // 4) segment mean (zero for isolated nodes).
    ecc_fin_kernel<<<(out_size + 255) / 256, 256, 0, stream>>>(out, counts, out_size);
}